// StatefulGPT2ForCausalLM_82669530513542
// MI455X (gfx1250) — compile-verified
//
#include <hip/hip_runtime.h>
#include <math.h>

#define LAYERS 12
#define BATCH  8
#define HEADS  12
#define CTXLEN 2048
#define HDIM   64
#define EMB    768
#define VOCAB  50257
#define CHUNK  256

typedef float v2f __attribute__((ext_vector_type(2)));
typedef float v8f __attribute__((ext_vector_type(8)));

static __device__ __forceinline__ v8f wmma4(v2f a, v2f b, v8f c) {
  // D(16x16,f32) = A(16x4,f32) x B(4x16,f32) + C
  return __builtin_amdgcn_wmma_f32_16x16x4_f32(false, a, false, b, (short)0, c, false, false);
}

static __device__ __forceinline__ v2f ld2(const float* p) {
  v2f r; r[0] = p[0]; r[1] = p[1]; return r;
}

static __device__ __forceinline__ float gelu_tanh(float x) {
  float x3 = x * x * x;
  return 0.5f * x * (1.f + tanhf(0.7978845608028654f * (x + 0.044715f * x3)));
}

// ---------------- embedding: h = wte[id] + wpe[pos] ----------------
__global__ void embed_kernel(const int* __restrict__ ids, const int* __restrict__ past_len,
                             const float* __restrict__ wte, const float* __restrict__ wpe,
                             float* __restrict__ h) {
  int idx = blockIdx.x * blockDim.x + threadIdx.x;   // [0, BATCH*EMB)
  int b = idx / EMB, e = idx % EMB;
  int tok = ids[b];
  int pos = past_len[0];
  h[idx] = wte[(size_t)tok * EMB + e] + wpe[(size_t)pos * EMB + e];
}

// ---------------- layernorm (one block per batch row) ----------------
__global__ __launch_bounds__(256)
void ln_kernel(const float* __restrict__ in, const float* __restrict__ g,
               const float* __restrict__ bta, float* __restrict__ out) {
  __shared__ float red[256];
  int b = blockIdx.x, tid = threadIdx.x;
  const float* row = in + (size_t)b * EMB;
  float s = 0.f;
  for (int i = tid; i < EMB; i += 256) s += row[i];
  red[tid] = s; __syncthreads();
  for (int o = 128; o > 0; o >>= 1) { if (tid < o) red[tid] += red[tid + o]; __syncthreads(); }
  float mean = red[0] * (1.f / EMB);
  __syncthreads();
  float v = 0.f;
  for (int i = tid; i < EMB; i += 256) { float d = row[i] - mean; v += d * d; }
  red[tid] = v; __syncthreads();
  for (int o = 128; o > 0; o >>= 1) { if (tid < o) red[tid] += red[tid + o]; __syncthreads(); }
  float rstd = rsqrtf(red[0] * (1.f / EMB) + 1e-5f);
  for (int i = tid; i < EMB; i += 256)
    out[(size_t)b * EMB + i] = (row[i] - mean) * rstd * g[i] + bta[i];
}

// ---------------- 8-row GEMM via V_WMMA_F32_16X16X4_F32 ----------------
// Out[b, j] = (RES ? Out[b,j] : 0) + act( bias[j] + sum_k X[b,k] * W(k,j) )
// TRANSB=false: W is (K x N) row-major, W(k,j) = W[k*N + j]
// TRANSB=true : W is (N x K) row-major, W(k,j) = W[j*K + k]   (logits: wte^T)
// K, N are compile-time so all weight loads get immediate offsets.
template <int ACT, bool RES, bool TRANSB, int K, int N>
__global__ __launch_bounds__(256)
void gemm8_kernel(const float* __restrict__ X, const float* __restrict__ W,
                  const float* __restrict__ bias, float* __restrict__ Out,
                  int Nvalid) {
  static_assert(CHUNK == 256 && (K % CHUNK) == 0, "chunking");
  __shared__ __align__(16) float sX[16 * CHUNK];     // rows 8..15 stay zero
  const int tid  = threadIdx.x;
  const int lane = tid & 31;
  const int wave = tid >> 5;
  const bool hi  = lane >= 16;
  const int m    = lane & 15;                         // A row held by this lane
  const int j    = blockIdx.x * 128 + wave * 16 + (lane & 15);
  const int jcl  = (j < Nvalid) ? j : (Nvalid - 1);
  const int koff = hi ? 2 : 0;                        // hi half carries K+2/K+3
  constexpr size_t sN = (size_t)N;

  // zero A rows 8..15 once (WMMA padding rows)
  {
    float4* z4 = (float4*)(sX + 8 * CHUNK);
    for (int q = tid; q < (8 * CHUNK) / 4; q += 256) z4[q] = make_float4(0.f, 0.f, 0.f, 0.f);
  }

  v8f acc = {0.f, 0.f, 0.f, 0.f, 0.f, 0.f, 0.f, 0.f};
  const float* srow = sX + m * CHUNK + koff;
  const float* wp = W + (size_t)koff * sN + j;        // !TRANSB running pointer
  const float* wq = W + (size_t)jcl * (size_t)K + koff; // TRANSB running pointer

  for (int k0 = 0; k0 < K; k0 += CHUNK) {
    __syncthreads();
    {   // stage 8 x CHUNK activations, float4-vectorized
      float4* s4 = (float4*)sX;
      for (int q = tid; q < (8 * CHUNK) / 4; q += 256) {
        int r = q >> 6;                               // 64 float4 per row
        int c = q & 63;
        s4[q] = *(const float4*)(X + (size_t)r * K + k0 + c * 4);
      }
    }
    __syncthreads();
#pragma unroll 1
    for (int kk = 0; kk < CHUNK; kk += 16) {          // 4 WMMAs per iteration
      const float* ap = srow + kk;
      v2f a0 = ld2(ap);      v2f a1 = ld2(ap + 4);
      v2f a2 = ld2(ap + 8);  v2f a3 = ld2(ap + 12);
      v2f b0, b1, b2, b3;
      if (TRANSB) {
        b0 = ld2(wq);      b1 = ld2(wq + 4);
        b2 = ld2(wq + 8);  b3 = ld2(wq + 12);
        wq += 16;
      } else {
        b0[0] = wp[0];        b0[1] = wp[sN];
        b1[0] = wp[4 * sN];   b1[1] = wp[5 * sN];
        b2[0] = wp[8 * sN];   b2[1] = wp[9 * sN];
        b3[0] = wp[12 * sN];  b3[1] = wp[13 * sN];
        wp += 16 * sN;
      }
      acc = wmma4(a0, b0, acc);
      acc = wmma4(a1, b1, acc);
      acc = wmma4(a2, b2, acc);
      acc = wmma4(a3, b3, acc);
    }
  }

  // lanes 0..15 hold D rows 0..7 in acc[0..7]
  if (lane < 16 && j < Nvalid) {
    float bb = bias ? bias[j] : 0.f;
#pragma unroll
    for (int r = 0; r < 8; ++r) {
      float v = acc[r] + bb;
      if (ACT == 1) v = gelu_tanh(v);
      if (RES) v += Out[(size_t)r * sN + j];
      Out[(size_t)r * sN + j] = v;
    }
  }
}

// ---------------- scatter new k/v into output caches at row past_len ----------------
__global__ void kv_scatter_kernel(const float* __restrict__ qkv, const int* __restrict__ past_len,
                                  float* __restrict__ outK, float* __restrict__ outV, int layer) {
  int idx = blockIdx.x * blockDim.x + threadIdx.x;   // [0, BATCH*EMB)
  int b = idx / EMB;
  int r = idx % EMB;                                 // hh*64 + d
  int pl = past_len[0];
  size_t base = ((((size_t)layer * BATCH + b) * HEADS + (r >> 6)) * CTXLEN + pl) * HDIM + (r & 63);
  outK[base] = qkv[(size_t)b * 3 * EMB + EMB + r];
  outV[base] = qkv[(size_t)b * 3 * EMB + 2 * EMB + r];
}

// ---------------- single-query attention over the updated cache ----------------
__global__ __launch_bounds__(256)
void attn_kernel(const float* __restrict__ qkv, const float* __restrict__ Kc,
                 const float* __restrict__ Vc, const int* __restrict__ past_len,
                 float* __restrict__ outO, int layer) {
  __shared__ float sS[CTXLEN];
  __shared__ __align__(16) float sq[HDIM];
  __shared__ float red[256];
  __shared__ float sO[4 * HDIM];
  int tid = threadIdx.x;
  int b  = blockIdx.x / HEADS;
  int hh = blockIdx.x % HEADS;
  int T  = past_len[0] + 1;    // single new token, fully causal-visible
  if (tid < HDIM) sq[tid] = qkv[(size_t)b * 3 * EMB + hh * HDIM + tid];
  __syncthreads();
  size_t base = (((size_t)layer * BATCH + b) * HEADS + hh) * (size_t)CTXLEN * HDIM;
  const float* Kb = Kc + base;
  const float* Vb = Vc + base;

  float lmax = -3.0e38f;
  for (int t = tid; t < T; t += 256) {
    const float4* kr = (const float4*)(Kb + (size_t)t * HDIM);
    const float4* qr = (const float4*)sq;
    float d = 0.f;
#pragma unroll
    for (int i = 0; i < 16; ++i) {
      float4 kv = kr[i], qv = qr[i];
      d += kv.x * qv.x + kv.y * qv.y + kv.z * qv.z + kv.w * qv.w;
    }
    d *= 0.125f;                 // 1/sqrt(64)
    sS[t] = d;
    lmax = fmaxf(lmax, d);
  }
  red[tid] = lmax; __syncthreads();
  for (int o = 128; o > 0; o >>= 1) { if (tid < o) red[tid] = fmaxf(red[tid], red[tid + o]); __syncthreads(); }
  float mx = red[0]; __syncthreads();

  float lsum = 0.f;
  for (int t = tid; t < T; t += 256) { float e = __expf(sS[t] - mx); sS[t] = e; lsum += e; }
  red[tid] = lsum; __syncthreads();
  for (int o = 128; o > 0; o >>= 1) { if (tid < o) red[tid] += red[tid + o]; __syncthreads(); }
  float denom = red[0];
  __syncthreads();

  int dd  = tid & 63;
  int grp = tid >> 6;
  float oacc = 0.f;
  for (int t = grp; t < T; t += 4) oacc += sS[t] * Vb[(size_t)t * HDIM + dd];  // coalesced in dd
  sO[grp * HDIM + dd] = oacc;
  __syncthreads();
  if (tid < HDIM) {
    float r = sO[tid] + sO[HDIM + tid] + sO[2 * HDIM + tid] + sO[3 * HDIM + tid];
    outO[(size_t)b * EMB + hh * HDIM + tid] = r / denom;
  }
}

extern "C" void kernel_launch(void* const* d_in, const int* in_sizes, int n_in,
                              void* d_out, int out_size, void* d_ws, size_t ws_size,
                              hipStream_t stream) {
  (void)in_sizes; (void)n_in; (void)out_size; (void)ws_size;
  const int*   input_ids  = (const int*)  d_in[0];
  /* d_in[1] attention_mask: unused by the reference math */
  const float* keyCache   = (const float*)d_in[2];
  const float* valueCache = (const float*)d_in[3];
  const int*   past_len   = (const int*)  d_in[4];
  const float* wte   = (const float*)d_in[5];
  const float* wpe   = (const float*)d_in[6];
  const float* ln1_g = (const float*)d_in[7];
  const float* ln1_b = (const float*)d_in[8];
  const float* attn_w= (const float*)d_in[9];
  const float* attn_b= (const float*)d_in[10];
  const float* proj_w= (const float*)d_in[11];
  const float* proj_b= (const float*)d_in[12];
  const float* ln2_g = (const float*)d_in[13];
  const float* ln2_b = (const float*)d_in[14];
  const float* fc_w  = (const float*)d_in[15];
  const float* fc_b  = (const float*)d_in[16];
  const float* fc2_w = (const float*)d_in[17];
  const float* fc2_b = (const float*)d_in[18];
  const float* lnf_g = (const float*)d_in[19];
  const float* lnf_b = (const float*)d_in[20];

  float* logits = (float*)d_out;
  const size_t cacheElems = (size_t)LAYERS * BATCH * HEADS * CTXLEN * HDIM;
  float* outK = logits + (size_t)BATCH * VOCAB;
  float* outV = outK + cacheElems;
  hipMemcpyAsync(outK, keyCache,   cacheElems * sizeof(float), hipMemcpyDeviceToDevice, stream);
  hipMemcpyAsync(outV, valueCache, cacheElems * sizeof(float), hipMemcpyDeviceToDevice, stream);

  float* h   = (float*)d_ws;
  float* x   = h   + BATCH * EMB;
  float* qkv = x   + BATCH * EMB;
  float* att = qkv + BATCH * 3 * EMB;
  float* ffn = att + BATCH * EMB;

  embed_kernel<<<(BATCH * EMB) / 256, 256, 0, stream>>>(input_ids, past_len, wte, wpe, h);

  for (int l = 0; l < LAYERS; ++l) {
    ln_kernel<<<BATCH, 256, 0, stream>>>(h, ln1_g + l * EMB, ln1_b + l * EMB, x);
    gemm8_kernel<0, false, false, EMB, 3 * EMB><<<(3 * EMB) / 128, 256, 0, stream>>>(
        x, attn_w + (size_t)l * EMB * 3 * EMB, attn_b + (size_t)l * 3 * EMB, qkv, 3 * EMB);
    kv_scatter_kernel<<<(BATCH * EMB) / 256, 256, 0, stream>>>(qkv, past_len, outK, outV, l);
    attn_kernel<<<BATCH * HEADS, 256, 0, stream>>>(qkv, outK, outV, past_len, att, l);
    gemm8_kernel<0, true, false, EMB, EMB><<<EMB / 128, 256, 0, stream>>>(
        att, proj_w + (size_t)l * EMB * EMB, proj_b + (size_t)l * EMB, h, EMB);
    ln_kernel<<<BATCH, 256, 0, stream>>>(h, ln2_g + l * EMB, ln2_b + l * EMB, x);
    gemm8_kernel<1, false, false, EMB, 4 * EMB><<<(4 * EMB) / 128, 256, 0, stream>>>(
        x, fc_w + (size_t)l * EMB * 4 * EMB, fc_b + (size_t)l * 4 * EMB, ffn, 4 * EMB);
    gemm8_kernel<0, true, false, 4 * EMB, EMB><<<EMB / 128, 256, 0, stream>>>(
        ffn, fc2_w + (size_t)l * 4 * EMB * EMB, fc2_b + (size_t)l * EMB, h, EMB);
  }

  ln_kernel<<<BATCH, 256, 0, stream>>>(h, lnf_g, lnf_b, x);
  gemm8_kernel<0, false, true, EMB, VOCAB><<<(VOCAB + 127) / 128, 256, 0, stream>>>(
      x, wte, nullptr, logits, VOCAB);
}